// Model_50697793962859
// MI455X (gfx1250) — compile-verified
//
#include <hip/hip_runtime.h>

typedef __attribute__((ext_vector_type(2))) float v2f;
typedef __attribute__((ext_vector_type(8))) float v8f;

#define HIDDEN 2048
#define GATES  (3 * HIDDEN)   // 6144
#define LAYERS 6
#define VOCAB  53

// ---------------------------------------------------------------------------
// x = emb[input[0]]
// ---------------------------------------------------------------------------
__global__ void embed_kernel(const int* __restrict__ input,
                             const float* __restrict__ emb,
                             float* __restrict__ xbuf) {
    int i = blockIdx.x * blockDim.x + threadIdx.x;
    if (i < HIDDEN) xbuf[i] = emb[(size_t)input[0] * HIDDEN + i];
}

// ---------------------------------------------------------------------------
// Combined gi / gh matvec for one layer.
//   blockIdx.x in [0, 768): 16 rows of the combined 12288-row output.
//   rows [0,6144)    : gi = w_ih_l @ x + b_ih_l
//   rows [6144,12288): gh = w_hh_l @ h + b_hh_l
// 8 waves per block split K=2048 into 256-wide slices; each wave accumulates a
// 16x16 f32 tile with V_WMMA_F32_16X16X4_F32 where the B matrix holds the
// vector broadcast into all 16 columns.  Deterministic LDS reduction.
// ---------------------------------------------------------------------------
__global__ void gru_matvec_kernel(const float* __restrict__ w_ih_l,
                                  const float* __restrict__ w_hh_l,
                                  const float* __restrict__ b_ih_l,
                                  const float* __restrict__ b_hh_l,
                                  const float* __restrict__ x,
                                  const float* __restrict__ h_l,
                                  float* __restrict__ gi,
                                  float* __restrict__ gh) {
    __shared__ float partial[8][16];

    const int tid  = threadIdx.x;
    const int lane = tid & 31;
    const int wave = tid >> 5;

    const int crow0 = blockIdx.x * 16;            // combined row base
    const bool isIH = (crow0 < GATES);
    const float* __restrict__ W    = isIH ? w_ih_l : w_hh_l;
    const float* __restrict__ v    = isIH ? x      : h_l;
    const float* __restrict__ bias = isIH ? b_ih_l : b_hh_l;
    float* __restrict__ out        = isIH ? gi     : gh;
    const int r0 = isIH ? crow0 : (crow0 - GATES);

    // A fragment (16x4 f32): lane L<16 -> row L, K={0,1}; lane L+16 -> row L, K={2,3}
    const int m  = lane & 15;
    const int kh = lane >> 4;
    const int kbase = wave * (HIDDEN / 8) + 2 * kh;   // 256-wide K slice per wave

    const float* __restrict__ wrow = W + (size_t)(r0 + m) * HIDDEN + kbase;
    const float* __restrict__ vv   = v + kbase;

    v8f c = {0.f, 0.f, 0.f, 0.f, 0.f, 0.f, 0.f, 0.f};

#pragma unroll 8
    for (int k = 0; k < HIDDEN / 8; k += 4) {
        v2f a = *(const v2f*)(wrow + k);   // W[row, k+2kh .. k+2kh+1]
        v2f b = *(const v2f*)(vv + k);     // v[k+2kh .. k+2kh+1] (all N columns equal)
        // D = A(16x4) x B(4x16) + C  ->  v_wmma_f32_16x16x4_f32
        c = __builtin_amdgcn_wmma_f32_16x16x4_f32(
                /*neg_a=*/false, a, /*neg_b=*/false, b,
                /*c_mod=*/(short)0, c, /*reuse_a=*/false, /*reuse_b=*/false);
    }

    // Column 0 of D: lane 0 holds rows 0..7 (c[0..7]), lane 16 holds rows 8..15.
    if (lane == 0) {
#pragma unroll
        for (int i = 0; i < 8; ++i) partial[wave][i] = c[i];
    } else if (lane == 16) {
#pragma unroll
        for (int i = 0; i < 8; ++i) partial[wave][8 + i] = c[i];
    }
    __syncthreads();

    if (wave == 0 && lane < 16) {
        float s = 0.f;
#pragma unroll
        for (int w = 0; w < 8; ++w) s += partial[w][lane];   // fixed order: deterministic
        out[r0 + lane] = s + bias[r0 + lane];
    }
}

// ---------------------------------------------------------------------------
// GRU gate math: h_new = (1-z)*n + z*h ; writes h_new to output slot and to
// xbuf (input of next layer / decode).
// ---------------------------------------------------------------------------
__global__ void gru_gate_kernel(const float* __restrict__ gi,
                                const float* __restrict__ gh,
                                const float* __restrict__ hprev,
                                float* __restrict__ hnew_out,
                                float* __restrict__ xbuf) {
    int i = blockIdx.x * blockDim.x + threadIdx.x;
    if (i >= HIDDEN) return;
    float ir = gi[i],              hr = gh[i];
    float iz = gi[HIDDEN + i],     hz = gh[HIDDEN + i];
    float in_ = gi[2 * HIDDEN + i], hn = gh[2 * HIDDEN + i];
    float r = 1.f / (1.f + expf(-(ir + hr)));
    float z = 1.f / (1.f + expf(-(iz + hz)));
    float n = tanhf(in_ + r * hn);
    float h = hprev[i];
    float hnew = (1.f - z) * n + z * h;
    hnew_out[i] = hnew;
    xbuf[i] = hnew;
}

// ---------------------------------------------------------------------------
// logits = w_dec @ x_out + b_dec  (53 x 2048): one wave per row.
// ---------------------------------------------------------------------------
__global__ void decode_kernel(const float* __restrict__ w_dec,
                              const float* __restrict__ b_dec,
                              const float* __restrict__ x,
                              float* __restrict__ logits) {
    int lane = threadIdx.x & 31;
    int wave = threadIdx.x >> 5;
    int row = blockIdx.x * (blockDim.x >> 5) + wave;
    if (row >= VOCAB) return;
    const float* wr = w_dec + (size_t)row * HIDDEN;
    float s = 0.f;
    for (int k = lane; k < HIDDEN; k += 32) s += wr[k] * x[k];
#pragma unroll
    for (int off = 16; off > 0; off >>= 1) s += __shfl_down(s, off, 32);
    if (lane == 0) logits[row] = s + b_dec[row];
}

// ---------------------------------------------------------------------------
extern "C" void kernel_launch(void* const* d_in, const int* in_sizes, int n_in,
                              void* d_out, int out_size, void* d_ws, size_t ws_size,
                              hipStream_t stream) {
    const int*   input  = (const int*)d_in[0];
    const float* hidden = (const float*)d_in[1];   // (6, 1, 2048)
    const float* emb    = (const float*)d_in[2];   // (53, 2048)
    const float* w_ih   = (const float*)d_in[3];   // (6, 6144, 2048)
    const float* w_hh   = (const float*)d_in[4];   // (6, 6144, 2048)
    const float* b_ih   = (const float*)d_in[5];   // (6, 6144)
    const float* b_hh   = (const float*)d_in[6];   // (6, 6144)
    const float* w_dec  = (const float*)d_in[7];   // (53, 2048)
    const float* b_dec  = (const float*)d_in[8];   // (53,)

    float* out  = (float*)d_out;       // [53 logits][6*2048 h_new]
    float* xbuf = (float*)d_ws;        // 2048 floats
    float* gi   = xbuf + HIDDEN;       // 6144 floats
    float* gh   = gi + GATES;          // 6144 floats

    embed_kernel<<<(HIDDEN + 255) / 256, 256, 0, stream>>>(input, emb, xbuf);

    for (int l = 0; l < LAYERS; ++l) {
        const float* w_ih_l = w_ih + (size_t)l * GATES * HIDDEN;
        const float* w_hh_l = w_hh + (size_t)l * GATES * HIDDEN;
        const float* b_ih_l = b_ih + (size_t)l * GATES;
        const float* b_hh_l = b_hh + (size_t)l * GATES;
        const float* h_l    = hidden + (size_t)l * HIDDEN;

        gru_matvec_kernel<<<(2 * GATES) / 16, 256, 0, stream>>>(
            w_ih_l, w_hh_l, b_ih_l, b_hh_l, xbuf, h_l, gi, gh);

        gru_gate_kernel<<<(HIDDEN + 255) / 256, 256, 0, stream>>>(
            gi, gh, h_l, out + VOCAB + (size_t)l * HIDDEN, xbuf);
    }

    decode_kernel<<<(VOCAB + 7) / 8, 256, 0, stream>>>(w_dec, b_dec, xbuf, out);
}